// one_layer_net_60129542144257
// MI455X (gfx1250) — compile-verified
//
#include <hip/hip_runtime.h>
#include <hip/hip_bf16.h>
#include <stdint.h>

#define STEPS   130
#define NW      108                 // 9*3*4 weights
#define NELEM   (STEPS * NW)        // 14040 noise / transformed-weight floats
#define CHUNKS  (NELEM / 4)         // 3510 float4 slots (async b128 chunks)
#define NTHREADS 256

// Gather index tables (flat index 0..28), lane = t*3 + j.
// j=0: A=x1=IDX1[t], B=x2=IDX2[t];  j=1: A=x0=IDX0[t], B=x2;  j=2: A=x0, B=x1.
__device__ const unsigned char kIdxA[32] = {
     1,  3,  3,   7,  0,  0,   4,  6,  6,  19, 12, 12,
    16,  9,  9,  13, 15, 15,  10, 18, 18,  22,  6,  6,
    25, 15, 15,  27, 28, 29, 30, 31 };
__device__ const unsigned char kIdxB[32] = {
    17, 17,  1,   5,  5,  7,  11, 11,  4,   8,  8, 19,
    28, 28, 16,   2,  2, 13,  27, 27, 10,  23, 23, 22,
    26, 26, 25,  27, 28, 29, 30, 31 };

__global__ __launch_bounds__(NTHREADS)
void triangle_scan_kernel(const float* __restrict__ x,
                          const float* __restrict__ weights,
                          const float* __restrict__ noise,
                          float* __restrict__ out) {
    // +NW pad: lets the scan prefetch s+1 unconditionally at s=129 (dead read).
    __shared__ __align__(16) float  s_w[NELEM + NW];
    __shared__ __align__(16) float  s_wt[NW];
    __shared__ float4               s_fixv[5];    // lanes 27..31 fixed-point slots
    const int tid = threadIdx.x;

    // ---- Phase 1: async DMA entire noise tensor (56160 B) global -> LDS ----
    for (int c = tid; c < CHUNKS; c += NTHREADS) {
        uint32_t lds_addr = (uint32_t)(uintptr_t)(&s_w[c * 4]);  // low 32b = LDS byte addr
        uint64_t gaddr    = (uint64_t)(uintptr_t)(noise + c * 4);
        asm volatile("global_load_async_to_lds_b128 %0, %1, off"
                     :: "v"(lds_addr), "v"(gaddr) : "memory");
    }
    if (tid < NW) s_wt[tid] = weights[tid];   // overlap with async DMA in flight
    asm volatile("s_wait_asynccnt 0" ::: "memory");
    __syncthreads();

    // ---- Phase 2 (parallel, off-chain): perturb weights and pre-factor mux ----
    // w_k = wt_k + |1-|wt_k||*n_k*0.125
    // store {0.25(w0+w1), 0.25(w1-w0), 0.25(w2+w3), 0.25(w3-w2)} per (step,lane)
    {
        float4*       s_w4  = reinterpret_cast<float4*>(s_w);
        const float4* s_wt4 = reinterpret_cast<const float4*>(s_wt);
        for (int f = tid; f < CHUNKS; f += NTHREADS) {
            int    l  = f % 27;                 // float4 index = s*27 + l
            float4 wt = s_wt4[l];
            float4 n  = s_w4[f];
            float w0 = wt.x + fabsf(1.0f - fabsf(wt.x)) * n.x * 0.125f;
            float w1 = wt.y + fabsf(1.0f - fabsf(wt.y)) * n.y * 0.125f;
            float w2 = wt.z + fabsf(1.0f - fabsf(wt.z)) * n.z * 0.125f;
            float w3 = wt.w + fabsf(1.0f - fabsf(wt.w)) * n.w * 0.125f;
            float4 r;
            r.x = 0.25f * (w0 + w1);
            r.y = 0.25f * (w1 - w0);
            r.z = 0.25f * (w2 + w3);
            r.w = 0.25f * (w3 - w2);
            s_w4[f] = r;
        }
    }
    __syncthreads();

    // ---- Phase 3: wave 0 runs the 130-step recurrence (lanes = flat state) ----
    if (tid < 32) {
        const int lane = tid;
        const int ia = (int)kIdxA[lane] * 4;    // ds_bpermute takes byte index
        const int ib = (int)kIdxB[lane] * 4;

        const float a = x[0];
        const float b = x[1];
        float state = -1.0f;                    // flat[0..26] init to -1
        if (lane == 27) state = a;              // flat[27] = x[0]
        if (lane == 28) state = b;              // flat[28] = x[1]
        if (lane > 28)  state = 0.0f;

        // Fixed P/Q slot {state/2, 0, state/2, 0} makes new == state exactly,
        // so the loop needs no lane<27 select on the critical path.
        if (lane >= 27) {
            float h = 0.5f * state;
            s_fixv[lane - 27] = make_float4(h, 0.0f, h, 0.0f);
        }

        const float4* wp;
        int stride;                              // in float4 units
        if (lane < 27) { wp = reinterpret_cast<const float4*>(s_w) + lane; stride = 27; }
        else           { wp = &s_fixv[lane - 27];                          stride = 0;  }

        float4 wv = *wp;                         // step-0 weights
        wp += stride;
        #pragma unroll 2
        for (int s = 0; s < STEPS; ++s) {
            // critical-path gathers
            float va = __uint_as_float((unsigned)__builtin_amdgcn_ds_bpermute(
                           ia, (int)__float_as_uint(state)));
            float vb = __uint_as_float((unsigned)__builtin_amdgcn_ds_bpermute(
                           ib, (int)__float_as_uint(state)));
            float4 wc = wv;
            wv = *wp;                            // prefetch s+1 (pad covers s=129)
            wp += stride;
            // 3 VALU levels between bpermutes:
            float m0 = __builtin_fmaf(va, wc.y, wc.x);   // 0.5*mux(A,w0,w1)
            float m1 = __builtin_fmaf(va, wc.w, wc.z);   // 0.5*mux(A,w2,w3)
            state    = __builtin_fmaf(vb, m1 - m0, m0 + m1);  // mux(B,m0,m1)
        }

        // outputs: final[0,1] -> lane 1, final[1,2] -> lane 5, final[7,2] -> lane 23
        if (lane == 1)  out[0] = state;
        if (lane == 5)  out[1] = state;
        if (lane == 23) out[2] = state;
    }
}

extern "C" void kernel_launch(void* const* d_in, const int* in_sizes, int n_in,
                              void* d_out, int out_size, void* d_ws, size_t ws_size,
                              hipStream_t stream) {
    (void)in_sizes; (void)n_in; (void)out_size; (void)d_ws; (void)ws_size;
    const float* x       = (const float*)d_in[0];   // [2]
    const float* weights = (const float*)d_in[1];   // [9,3,4] = 108
    const float* noise   = (const float*)d_in[2];   // [130,9,3,4] = 14040
    float* out           = (float*)d_out;           // [3]
    hipLaunchKernelGGL(triangle_scan_kernel, dim3(1), dim3(NTHREADS), 0, stream,
                       x, weights, noise, out);
}